// HoiOutputLayers_79431125172974
// MI455X (gfx1250) — compile-verified
//
#include <hip/hip_runtime.h>
#include <hip/hip_bf16.h>
#include <stdint.h>

#define SCORE_THRESH 0.05f
#define NBUCK 4096     // histogram over (positive) float bits >> 19
#define CAP   4096     // candidate buffer size
#define T1    256      // streaming block size (8 waves on wave32)

// ws layout (32-bit words):
//  [0]                          candidate counter
//  [1]                          threshold bits T
//  [16 .. 16+NBUCK)             global histogram
//  [16+NBUCK .. 16+NBUCK+CAP)   candidate score bits
//  [16+NBUCK+CAP .. +2*CAP)     candidate flat indices

__global__ void hoi_init(unsigned* ws32) {
  unsigned i = blockIdx.x * blockDim.x + threadIdx.x;
  if (i < 16u + NBUCK) ws32[i] = 0u;
}

__global__ __launch_bounds__(T1) void hoi_pass1(
    const float* __restrict__ hoi, const float* __restrict__ ps,
    const float* __restrict__ os, unsigned* __restrict__ ghist,
    unsigned N, unsigned N4, unsigned K) {
  __shared__ unsigned hist[NBUCK];
  __shared__ __align__(16) float stage[T1 * 4];
  for (unsigned b = threadIdx.x; b < NBUCK; b += blockDim.x) hist[b] = 0u;
  __syncthreads();

  const unsigned ldsb = (unsigned)(size_t)&stage[threadIdx.x * 4];
  const unsigned stride = gridDim.x * blockDim.x;
  for (unsigned q = blockIdx.x * blockDim.x + threadIdx.x; q < N4; q += stride) {
    unsigned long long ga = (unsigned long long)(hoi + 4ull * q);
    // CDNA5 async global->LDS stage (per-lane self-staging; no barrier needed)
    asm volatile(
        "global_load_async_to_lds_b128 %0, %1, off\n\t"
        "s_wait_asynccnt 0"
        :: "v"(ldsb), "v"(ga) : "memory");
    float4 v = *reinterpret_cast<const float4*>(&stage[threadIdx.x * 4]);

    unsigned i0 = q * 4u;
    unsigned r0 = i0 / K;
    unsigned rem = i0 - r0 * K;
    float e[4] = {v.x, v.y, v.z, v.w};
#pragma unroll
    for (int j = 0; j < 4; ++j) {
      unsigned r = r0 + ((rem + (unsigned)j) >= K ? 1u : 0u);  // K >= 4
      float sc = e[j] * ps[r] * os[r];
      if (sc > SCORE_THRESH) {
        unsigned bkt = __float_as_uint(sc) >> 19;  // positive -> monotone, < 4096
        atomicAdd(&hist[bkt], 1u);
      }
    }
  }
  // scalar tail (N not multiple of 4)
  {
    unsigned gid = blockIdx.x * blockDim.x + threadIdx.x;
    unsigned base = N4 * 4u;
    if (base + gid < N) {
      unsigned i = base + gid;
      unsigned r = i / K;
      float sc = hoi[i] * ps[r] * os[r];
      if (sc > SCORE_THRESH) atomicAdd(&hist[__float_as_uint(sc) >> 19], 1u);
    }
  }
  __syncthreads();
  for (unsigned b = threadIdx.x; b < NBUCK; b += blockDim.x) {
    unsigned c = hist[b];
    if (c) atomicAdd(&ghist[b], c);
  }
}

__global__ __launch_bounds__(1024) void hoi_scan(
    const unsigned* __restrict__ ghist, unsigned* __restrict__ ctrl, int topk) {
  __shared__ unsigned sa[NBUCK];
  __shared__ unsigned sb[NBUCK];
  __shared__ int best;
  if (threadIdx.x == 0) best = -1;
  for (unsigned i = threadIdx.x; i < NBUCK; i += blockDim.x) sa[i] = ghist[i];
  __syncthreads();
  unsigned* src = sa;
  unsigned* dst = sb;
  for (unsigned off = 1; off < NBUCK; off <<= 1) {   // suffix sums (Hillis-Steele)
    for (unsigned i = threadIdx.x; i < NBUCK; i += blockDim.x)
      dst[i] = src[i] + ((i + off < NBUCK) ? src[i + off] : 0u);
    __syncthreads();
    unsigned* t = src; src = dst; dst = t;
  }
  for (unsigned i = threadIdx.x; i < NBUCK; i += blockDim.x)
    if (src[i] >= (unsigned)topk) atomicMax(&best, (int)i);
  __syncthreads();
  if (threadIdx.x == 0)
    ctrl[1] = (best < 0) ? 0u : ((unsigned)best << 19);
}

__global__ __launch_bounds__(T1) void hoi_pass2(
    const float* __restrict__ hoi, const float* __restrict__ ps,
    const float* __restrict__ os, unsigned* __restrict__ ctrl,
    unsigned* __restrict__ cbits, unsigned* __restrict__ cidx,
    unsigned N, unsigned N4, unsigned K) {
  const unsigned T = ctrl[1];
  const unsigned stride = gridDim.x * blockDim.x;
  for (unsigned q = blockIdx.x * blockDim.x + threadIdx.x; q < N4; q += stride) {
    float4 v = *reinterpret_cast<const float4*>(hoi + 4ull * q);
    unsigned i0 = q * 4u;
    unsigned r0 = i0 / K;
    unsigned rem = i0 - r0 * K;
    float e[4] = {v.x, v.y, v.z, v.w};
#pragma unroll
    for (int j = 0; j < 4; ++j) {
      unsigned r = r0 + ((rem + (unsigned)j) >= K ? 1u : 0u);
      float sc = e[j] * ps[r] * os[r];
      unsigned bits = __float_as_uint(sc);
      if (sc > SCORE_THRESH && bits >= T) {
        unsigned pos = atomicAdd(&ctrl[0], 1u);
        if (pos < CAP) { cbits[pos] = bits; cidx[pos] = i0 + (unsigned)j; }
      }
    }
  }
  {
    unsigned gid = blockIdx.x * blockDim.x + threadIdx.x;
    unsigned base = N4 * 4u;
    if (base + gid < N) {
      unsigned i = base + gid;
      unsigned r = i / K;
      float sc = hoi[i] * ps[r] * os[r];
      unsigned bits = __float_as_uint(sc);
      if (sc > SCORE_THRESH && bits >= T) {
        unsigned pos = atomicAdd(&ctrl[0], 1u);
        if (pos < CAP) { cbits[pos] = bits; cidx[pos] = i; }
      }
    }
  }
}

__global__ __launch_bounds__(1024) void hoi_final(
    const unsigned* __restrict__ ctrl, const unsigned* __restrict__ cbits,
    const unsigned* __restrict__ cidx, const float* __restrict__ pboxes,
    const float* __restrict__ oboxes, const int* __restrict__ ocls,
    float* __restrict__ out, int topk, unsigned K) {
  __shared__ unsigned long long keys[CAP];
  unsigned n = ctrl[0];
  if (n > CAP) n = CAP;
  // key = (score_bits << 32) | ~idx  -> ascending sort gives value desc / idx asc at the top
  for (unsigned i = threadIdx.x; i < CAP; i += blockDim.x)
    keys[i] = (i < n) ? ((((unsigned long long)cbits[i]) << 32) |
                         (unsigned long long)(0xFFFFFFFFu - cidx[i]))
                      : 0ull;
  __syncthreads();
  for (unsigned k = 2; k <= CAP; k <<= 1) {
    for (unsigned j = k >> 1; j > 0; j >>= 1) {
      for (unsigned i = threadIdx.x; i < CAP; i += blockDim.x) {
        unsigned ixj = i ^ j;
        if (ixj > i) {
          unsigned long long a = keys[i], b = keys[ixj];
          bool up = ((i & k) == 0);
          if ((a > b) == up) { keys[i] = b; keys[ixj] = a; }
        }
      }
      __syncthreads();
    }
  }
  // sorted ascending; t-th best at CAP-1-t
  for (int t = (int)threadIdx.x; t < topk; t += (int)blockDim.x) {
    bool valid = (unsigned)t < n;
    unsigned long long key = valid ? keys[CAP - 1 - t] : 0ull;
    unsigned bits = (unsigned)(key >> 32);
    unsigned idx = valid ? (0xFFFFFFFFu - (unsigned)(key & 0xFFFFFFFFull)) : 0u;
    float sc = valid ? __uint_as_float(bits) : 0.0f;
    unsigned pair = idx / K;
    unsigned act = idx - pair * K;
    out[t] = sc;                                        // scores
#pragma unroll
    for (int c = 0; c < 4; ++c) {
      out[topk + 4 * t + c]     = pboxes[4u * pair + (unsigned)c];   // person boxes
      out[5 * topk + 4 * t + c] = oboxes[4u * pair + (unsigned)c];   // object boxes
    }
    out[9 * topk + t]  = (float)ocls[pair];             // object classes
    out[10 * topk + t] = valid ? (float)act : 0.0f;     // action classes
    out[11 * topk + t] = valid ? 1.0f : 0.0f;           // valid
  }
}

extern "C" void kernel_launch(void* const* d_in, const int* in_sizes, int n_in,
                              void* d_out, int out_size, void* d_ws, size_t ws_size,
                              hipStream_t stream) {
  (void)n_in; (void)ws_size;
  const float* pboxes = (const float*)d_in[0];
  const float* oboxes = (const float*)d_in[1];
  const float* ps     = (const float*)d_in[2];
  const float* os     = (const float*)d_in[3];
  const int*   ocls   = (const int*)d_in[4];
  const float* hoi    = (const float*)d_in[5];
  float* out = (float*)d_out;

  unsigned R  = (unsigned)in_sizes[2];        // person_box_scores length
  unsigned N  = (unsigned)in_sizes[5];        // R*K flat
  unsigned K  = (R > 0u) ? (N / R) : 1u;
  unsigned N4 = N / 4u;
  int topk = out_size / 12;                   // 1+4+4+1+1+1 outputs per slot

  unsigned* ws32  = (unsigned*)d_ws;
  unsigned* ctrl  = ws32;
  unsigned* ghist = ws32 + 16;
  unsigned* cbits = ws32 + 16 + NBUCK;
  unsigned* cidx  = ws32 + 16 + NBUCK + CAP;

  hoi_init<<<(16 + NBUCK + 255) / 256, 256, 0, stream>>>(ws32);

  unsigned blocks = (N4 + T1 - 1) / T1;
  if (blocks > 1024u) blocks = 1024u;
  if (blocks == 0u) blocks = 1u;

  hoi_pass1<<<blocks, T1, 0, stream>>>(hoi, ps, os, ghist, N, N4, K);
  hoi_scan<<<1, 1024, 0, stream>>>(ghist, ctrl, topk);
  hoi_pass2<<<blocks, T1, 0, stream>>>(hoi, ps, os, ctrl, cbits, cidx, N, N4, K);
  hoi_final<<<1, 1024, 0, stream>>>(ctrl, cbits, cidx, pboxes, oboxes, ocls, out, topk, K);
}